// WordVanilla_86990267613906
// MI455X (gfx1250) — compile-verified
//
#include <hip/hip_runtime.h>
#include <math.h>

// WordVanilla RNN forward for MI455X (gfx1250, wave32).
//   e   = emb[x]                                [B,T,E]
//   h_t = tanh(e_t @ Wxh + h_{t-1} @ Whh + bh)  scan over T
//   out = softmax(h_T/(1-dr) @ Wd + bd)         [B,L]
// B=16384, T=128, E=H=20, L=15.
//
// One wave == one 16-row batch tile. fp32 WMMA 16x16x4, K=20 -> 5 chunks,
// N=20 -> 2 tiles (second zero-padded). Weights live in B-layout fragments
// in VGPRs for the whole scan; h is carried in A-layout fragments, with a
// per-wave LDS round trip to transpose the WMMA D-layout back to A-layout.
// tanh uses the CDNA5 hardware TRANS op (v_tanh_f32) when available.

typedef __attribute__((ext_vector_type(2))) float v2f;
typedef __attribute__((ext_vector_type(8))) float v8f;

#if defined(__has_builtin)
#if __has_builtin(__builtin_amdgcn_tanhf)
#define FAST_TANH(x) __builtin_amdgcn_tanhf(x)
#endif
#endif
#ifndef FAST_TANH
#define FAST_TANH(x) tanhf(x)
#endif

#define WAVES_PER_BLOCK 8
#define ROWS_PER_WAVE   16
#define T_STEPS         128
#define E_DIM           20
#define H_DIM           20
#define L_DIM           15
#define LDS_STRIDE      22   // floats; keeps m*stride + 4c + 2*half 8B-aligned

__global__ __launch_bounds__(256)
void WordVanilla_rnn_kernel(const int*   __restrict__ x,
                            const float* __restrict__ drop_rate,
                            const float* __restrict__ emb,
                            const float* __restrict__ Wxh,
                            const float* __restrict__ Whh,
                            const float* __restrict__ bh,
                            const float* __restrict__ Wd,
                            const float* __restrict__ bd,
                            float*       __restrict__ out,
                            int B)
{
    __shared__ float lds[WAVES_PER_BLOCK][ROWS_PER_WAVE][LDS_STRIDE];

    const int tid  = threadIdx.x;
    const int wave = tid >> 5;
    const int lane = tid & 31;
    const int half = lane >> 4;   // lane group: 0 => lanes 0-15, 1 => 16-31
    const int m    = lane & 15;   // A-frag row / B,D-frag column (low bits)
    const int tile = blockIdx.x * WAVES_PER_BLOCK + wave;
    const int b0   = tile * ROWS_PER_WAVE;
    if (b0 + ROWS_PER_WAVE > B) return;   // uniform per wave: EXEC stays all-1s

    // ---- preload Wxh/Whh as B-layout fragments (constant over the scan) ----
    // B 4x16 f32 chunk layout: VGPR v, lane-half h: K = 4c + v + 2h, N = m+16*nt
    //   => frag.x = W[4c+2*half][n], frag.y = W[4c+2*half+1][n]
    v2f wxh[2][5], whh[2][5];
    float bh_n[2];
#pragma unroll
    for (int nt = 0; nt < 2; ++nt) {
        const int  n  = m + 16 * nt;
        const bool nv = (n < H_DIM);
        bh_n[nt] = nv ? bh[n] : 0.0f;
#pragma unroll
        for (int c = 0; c < 5; ++c) {
            const int k = 4 * c + 2 * half;
            wxh[nt][c].x = nv ? Wxh[(k    ) * H_DIM + n] : 0.0f;
            wxh[nt][c].y = nv ? Wxh[(k + 1) * H_DIM + n] : 0.0f;
            whh[nt][c].x = nv ? Whh[(k    ) * H_DIM + n] : 0.0f;
            whh[nt][c].y = nv ? Whh[(k + 1) * H_DIM + n] : 0.0f;
        }
    }

    // ---- h state as A-layout fragments: lane(half,m) holds h[m][4c+2h+{0,1}]
    v2f hA[5];
#pragma unroll
    for (int c = 0; c < 5; ++c) { hA[c].x = 0.0f; hA[c].y = 0.0f; }

    const int*  xrow  = x + (size_t)(b0 + m) * T_STEPS;  // 512B-aligned row
    float*      myLds = &lds[wave][0][0];

    for (int t0 = 0; t0 < T_STEPS; t0 += 4) {
        // prefetch 4 tokens at once (pulls the token load off the per-step
        // critical path; xrow + t0 is 16B aligned)
        const int4 toks = *(const int4*)(xrow + t0);
#pragma unroll
        for (int tt = 0; tt < 4; ++tt) {
            const int tok = (tt == 0) ? toks.x : (tt == 1) ? toks.y
                          : (tt == 2) ? toks.z : toks.w;

            // gather embedding row fragments directly in A layout (8B aligned)
            const float* erow = emb + (size_t)tok * E_DIM + 2 * half;
            v2f eA[5];
#pragma unroll
            for (int c = 0; c < 5; ++c)
                eA[c] = *(const v2f*)(erow + 4 * c);

            // acc = e @ Wxh + h @ Whh  for two N tiles (fp32 WMMA 16x16x4)
            v8f acc0 = {}; v8f acc1 = {};
#pragma unroll
            for (int c = 0; c < 5; ++c)
                acc0 = __builtin_amdgcn_wmma_f32_16x16x4_f32(
                    false, eA[c], false, wxh[0][c], (short)0, acc0, false, false);
#pragma unroll
            for (int c = 0; c < 5; ++c)
                acc0 = __builtin_amdgcn_wmma_f32_16x16x4_f32(
                    false, hA[c], false, whh[0][c], (short)0, acc0, false, false);
#pragma unroll
            for (int c = 0; c < 5; ++c)
                acc1 = __builtin_amdgcn_wmma_f32_16x16x4_f32(
                    false, eA[c], false, wxh[1][c], (short)0, acc1, false, false);
#pragma unroll
            for (int c = 0; c < 5; ++c)
                acc1 = __builtin_amdgcn_wmma_f32_16x16x4_f32(
                    false, hA[c], false, whh[1][c], (short)0, acc1, false, false);

            // h_new = tanh(acc + bh); D layout: acc[i] => row i+8*half, col m(+16)
#pragma unroll
            for (int i = 0; i < 8; ++i) {
                const int row = i + 8 * half;
                myLds[row * LDS_STRIDE + m] = FAST_TANH(acc0[i] + bh_n[0]);
            }
            if (m < (H_DIM - 16)) {               // n = m+16 valid only for m<4
#pragma unroll
                for (int i = 0; i < 8; ++i) {
                    const int row = i + 8 * half;
                    myLds[row * LDS_STRIDE + m + 16] = FAST_TANH(acc1[i] + bh_n[1]);
                }
            }

            // transpose back: read h in A layout (same-wave LDS, DScnt-ordered)
#pragma unroll
            for (int c = 0; c < 5; ++c)
                hA[c] = *(const v2f*)(myLds + m * LDS_STRIDE + 4 * c + 2 * half);
        }
    }

    // ---- final dense + softmax: h_last already in LDS; one row per lane ----
    const float scale = 1.0f / (1.0f - drop_rate[0]);
    if (half == 0) {
        float hrow[H_DIM];
#pragma unroll
        for (int k = 0; k < H_DIM; ++k)
            hrow[k] = myLds[m * LDS_STRIDE + k] * scale;

        float z[L_DIM];
        float zmax = -INFINITY;
#pragma unroll
        for (int l = 0; l < L_DIM; ++l) {
            float s = bd[l];
#pragma unroll
            for (int k = 0; k < H_DIM; ++k)
                s += hrow[k] * Wd[k * L_DIM + l];
            z[l] = s;
            zmax = fmaxf(zmax, s);
        }
        float sum = 0.0f;
#pragma unroll
        for (int l = 0; l < L_DIM; ++l) {
            z[l] = __expf(z[l] - zmax);   // native v_exp_f32
            sum += z[l];
        }
        const float inv = 1.0f / sum;
        float* orow = out + (size_t)(b0 + m) * L_DIM;
#pragma unroll
        for (int l = 0; l < L_DIM; ++l)
            orow[l] = z[l] * inv;
    }
}

extern "C" void kernel_launch(void* const* d_in, const int* in_sizes, int n_in,
                              void* d_out, int out_size, void* d_ws, size_t ws_size,
                              hipStream_t stream) {
    (void)n_in; (void)out_size; (void)d_ws; (void)ws_size;

    const int*   x   = (const int*)  d_in[0];
    const float* dr  = (const float*)d_in[1];
    const float* emb = (const float*)d_in[2];
    const float* Wxh = (const float*)d_in[3];
    const float* Whh = (const float*)d_in[4];
    const float* bh  = (const float*)d_in[5];
    const float* Wd  = (const float*)d_in[6];
    const float* bd  = (const float*)d_in[7];
    float*       out = (float*)d_out;

    const int B = in_sizes[0] / T_STEPS;                       // 16384
    const int rowsPerBlock = ROWS_PER_WAVE * WAVES_PER_BLOCK;  // 128
    const int grid = (B + rowsPerBlock - 1) / rowsPerBlock;    // 128 blocks

    hipLaunchKernelGGL(WordVanilla_rnn_kernel, dim3(grid), dim3(256), 0, stream,
                       x, dr, emb, Wxh, Whh, bh, Wd, bd, out, B);
}